// LightGCN_4861902979538
// MI455X (gfx1250) — compile-verified
//
#include <hip/hip_runtime.h>
#include <hip/hip_bf16.h>
#include <math.h>

#define N_U   100000
#define N_I   50000
#define DIM   64
#define NNZ_E 2000000
#define BATCH 4096
#define KEEP_SCALE 1.25f      /* 1/(1-0.2) inverted dropout */
#define LAMBDA_1 1e-4f

typedef __attribute__((ext_vector_type(2))) float v2f;
typedef __attribute__((ext_vector_type(8))) float v8f;

// ---------------------------------------------------------------------------
// Zero workspace (Z buffers + accumulators). Grid-stride.
// ---------------------------------------------------------------------------
__global__ void lg_zero_f32(float* __restrict__ p, size_t n) {
    size_t i = (size_t)blockIdx.x * blockDim.x + threadIdx.x;
    size_t stride = (size_t)gridDim.x * blockDim.x;
    for (; i < n; i += stride) p[i] = 0.0f;
}

// ---------------------------------------------------------------------------
// Edge-parallel SpMM scatter: Z[dst] += w * E[src].  16 lanes per edge,
// one float4 (16B) per lane -> coalesced 256B row segments, f32 HW atomics.
// All matrices are L2-resident on MI455X (192MB L2), so gathers/atomics
// stay on-die; HBM only streams the edge arrays.
// ---------------------------------------------------------------------------
__global__ void lg_spmm_scatter(const float* __restrict__ vals,
                                const int*  __restrict__ src_idx,
                                const int*  __restrict__ dst_idx,
                                const unsigned char* __restrict__ keep,
                                const float* __restrict__ E_src,   // [n_src, 64]
                                float*       __restrict__ Z_dst,   // [n_dst, 64]
                                int nnz, float scale) {
    int t = blockIdx.x * blockDim.x + threadIdx.x;
    int e = t >> 4;          // edge id
    int c = t & 15;          // float4 column chunk within the 64-wide row
    if (e >= nnz) return;
    if (c == 0) {            // stream-ahead prefetch of edge arrays (gfx1250 global_prefetch_b8)
        __builtin_prefetch(vals + e + 4096, 0, 0);
        __builtin_prefetch(src_idx + e + 4096, 0, 0);
        __builtin_prefetch(dst_idx + e + 4096, 0, 0);
    }
    if (!keep[e]) return;    // sparse dropout: 20% of edges dropped
    float w = vals[e] * scale;
    int s = src_idx[e];
    int d = dst_idx[e];
    const float4 v = ((const float4*)E_src)[(size_t)s * 16 + c];
    float* zp = Z_dst + (size_t)d * DIM + c * 4;
    unsafeAtomicAdd(zp + 0, w * v.x);
    unsafeAtomicAdd(zp + 1, w * v.y);
    unsafeAtomicAdd(zp + 2, w * v.z);
    unsafeAtomicAdd(zp + 3, w * v.w);
}

// ---------------------------------------------------------------------------
// BPR score + loss partials via WMMA. One wave (32 lanes) per tile of 16
// triples. Builds U (16x64), P (16x64), N (16x64) tiles in LDS, computes
// diag(U*P^T) and diag(U*N^T) with V_WMMA_F32_16X16X4_F32 over K=64
// (16 MACC steps), then accumulates sum(log_sigmoid) and sum(squares).
// ---------------------------------------------------------------------------
__global__ __launch_bounds__(32)
void lg_score_wmma(const float* __restrict__ Eu0, const float* __restrict__ Ei0,
                   const float* __restrict__ Zu1, const float* __restrict__ Zu2,
                   const float* __restrict__ Zi1, const float* __restrict__ Zi2,
                   const int* __restrict__ uids, const int* __restrict__ pos,
                   const int* __restrict__ neg, float* __restrict__ acc) {
    __shared__ float ut[16][DIM];
    __shared__ float pt[16][DIM];
    __shared__ float nt[16][DIM];

    const int l    = threadIdx.x;     // 0..31
    const int i    = l & 15;          // triple within tile (== M == N index)
    const int half = l >> 4;          // lane half-group
    const int t0   = blockIdx.x * 16;

    const int u = uids[t0 + i];
    const int p = pos [t0 + i];
    const int n = neg [t0 + i];

    // Stage summed embeddings E0 + Z(layer1) + Z(layer2); half-group covers 32 cols.
    float ssq = 0.0f;
    const int cbase = half * 32;
    for (int c = 0; c < 32; ++c) {
        int col = cbase + c;
        float eu = Eu0[(size_t)u * DIM + col];
        float ep = Ei0[(size_t)p * DIM + col];
        float en = Ei0[(size_t)n * DIM + col];
        ssq += eu * eu + ep * ep + en * en;   // L2 reg uses layer-0 embeddings
        ut[i][col] = eu + Zu1[(size_t)u * DIM + col] + Zu2[(size_t)u * DIM + col];
        pt[i][col] = ep + Zi1[(size_t)p * DIM + col] + Zi2[(size_t)p * DIM + col];
        nt[i][col] = en + Zi1[(size_t)n * DIM + col] + Zi2[(size_t)n * DIM + col];
    }
    __syncthreads();

    // 16x16x4 f32 WMMA layout (wave32):
    //   A (16x4):  lane -> M=i, VGPR0/1 hold K = k0+2*half {+0,+1}
    //   B (4x16):  lane -> N=i, VGPR0/1 hold K = k0+2*half {+0,+1}; B[k][n]=P[n][k]
    v8f cp = {};
    v8f cn = {};
    for (int k0 = 0; k0 < DIM; k0 += 4) {
        int kk = k0 + half * 2;
        v2f a, bp, bn;
        a.x  = ut[i][kk]; a.y  = ut[i][kk + 1];
        bp.x = pt[i][kk]; bp.y = pt[i][kk + 1];
        bn.x = nt[i][kk]; bn.y = nt[i][kk + 1];
        cp = __builtin_amdgcn_wmma_f32_16x16x4_f32(false, a, false, bp, (short)0, cp, false, false);
        cn = __builtin_amdgcn_wmma_f32_16x16x4_f32(false, a, false, bn, (short)0, cn, false, false);
    }

    // Diagonal of C: rows 0..7 live at lanes 0..7 (c[lane]); rows 8..15 at
    // lanes 24..31 (c[lane-24]).
    float dp = 0.0f, dn = 0.0f;
    bool have = false;
    if (l < 8)        { dp = cp[l];      dn = cn[l];      have = true; }
    else if (l >= 24) { dp = cp[l - 24]; dn = cn[l - 24]; have = true; }
    if (have) {
        float x  = dp - dn;
        // log_sigmoid(x), numerically stable
        float ls = fminf(x, 0.0f) - log1pf(expf(-fabsf(x)));
        unsafeAtomicAdd(&acc[0], ls);
    }
    unsafeAtomicAdd(&acc[1], ssq);
}

// ---------------------------------------------------------------------------
// Final scalar combine.
// ---------------------------------------------------------------------------
__global__ void lg_finalize(const float* __restrict__ acc, float* __restrict__ out) {
    float loss_r   = -acc[0] / (float)BATCH;
    float loss_reg = 0.5f * acc[1] / (float)BATCH;
    out[0] = loss_r + LAMBDA_1 * loss_reg;   // loss
    out[1] = loss_r;                         // loss_r
}

// ---------------------------------------------------------------------------
extern "C" void kernel_launch(void* const* d_in, const int* in_sizes, int n_in,
                              void* d_out, int out_size, void* d_ws, size_t ws_size,
                              hipStream_t stream) {
    const float*         Eu0  = (const float*)d_in[0];          // [N_U, 64]
    const float*         Ei0  = (const float*)d_in[1];          // [N_I, 64]
    const float*         vals = (const float*)d_in[2];          // [NNZ]
    const int*           rows = (const int*)d_in[3];            // [NNZ]
    const int*           cols = (const int*)d_in[4];            // [NNZ]
    const unsigned char* keep = (const unsigned char*)d_in[5];  // [L,2,NNZ] bool
    const int*           uids = (const int*)d_in[6];            // [B]
    const int*           pos  = (const int*)d_in[7];            // [B]
    const int*           neg  = (const int*)d_in[8];            // [B]

    float* ws  = (float*)d_ws;
    float* Zu1 = ws;
    float* Zu2 = Zu1 + (size_t)N_U * DIM;
    float* Zi1 = Zu2 + (size_t)N_U * DIM;
    float* Zi2 = Zi1 + (size_t)N_I * DIM;
    float* acc = Zi2 + (size_t)N_I * DIM;   // [0]=sum log_sigmoid, [1]=sum squares

    size_t ztotal = 2 * (size_t)N_U * DIM + 2 * (size_t)N_I * DIM + 2;
    lg_zero_f32<<<8192, 256, 0, stream>>>(ws, ztotal);

    const int spmm_blocks = (NNZ_E * 16 + 255) / 256;   // 16 lanes per edge
    // Layer 1: Z_u1 = drop(A)  @ E_i0 ; Z_i1 = drop(A)^T @ E_u0
    lg_spmm_scatter<<<spmm_blocks, 256, 0, stream>>>(
        vals, cols, rows, keep + 0 * (size_t)NNZ_E, Ei0, Zu1, NNZ_E, KEEP_SCALE);
    lg_spmm_scatter<<<spmm_blocks, 256, 0, stream>>>(
        vals, rows, cols, keep + 1 * (size_t)NNZ_E, Eu0, Zi1, NNZ_E, KEEP_SCALE);
    // Layer 2: Z_u2 = drop(A)  @ Z_i1 ; Z_i2 = drop(A)^T @ Z_u1
    lg_spmm_scatter<<<spmm_blocks, 256, 0, stream>>>(
        vals, cols, rows, keep + 2 * (size_t)NNZ_E, Zi1, Zu2, NNZ_E, KEEP_SCALE);
    lg_spmm_scatter<<<spmm_blocks, 256, 0, stream>>>(
        vals, rows, cols, keep + 3 * (size_t)NNZ_E, Zu1, Zi2, NNZ_E, KEEP_SCALE);

    // BPR scores + loss partials via WMMA (one wave per 16 triples).
    lg_score_wmma<<<BATCH / 16, 32, 0, stream>>>(Eu0, Ei0, Zu1, Zu2, Zi1, Zi2,
                                                 uids, pos, neg, acc);
    lg_finalize<<<1, 1, 0, stream>>>(acc, (float*)d_out);
}